// GCN_1layer_6889127543165
// MI455X (gfx1250) — compile-verified
//
#include <hip/hip_runtime.h>

typedef __attribute__((ext_vector_type(2))) float v2f;
typedef __attribute__((ext_vector_type(8))) float v8f;

#define D_FEAT 128
#define ROWS_PER_WAVE 16
#define WAVES_PER_BLOCK 4           // 128 threads (4 x wave32)
#define LDS_STRIDE 132              // 132 mod 64 = 4 -> conflict-free A reads

// ---------------- K0: zero degree counters ----------------
__global__ void gcn_zero_u32(unsigned* __restrict__ p, int n) {
    int i = blockIdx.x * blockDim.x + threadIdx.x;
    if (i < n) p[i] = 0u;
}

// ---------------- K1: in-degree of dst (edges only; +1 self-loop later) ----
__global__ void gcn_degree(const int* __restrict__ dst, unsigned* __restrict__ deg, int E) {
    int e = blockIdx.x * blockDim.x + threadIdx.x;
    if (e < E) atomicAdd(&deg[dst[e]], 1u);
}

// ---------------- K2: h = x @ W via V_WMMA_F32_16X16X4_F32 ----------------
// One wave computes 16 rows of h. The 16x128 tile of x is staged in LDS
// (coalesced float4 global loads), then consumed in the 16x4 f32 A-fragment
// layout: lanes 0-15 hold M=lane with K={0,1}, lanes 16-31 hold M=lane-16
// with K={2,3}. B (4x16) carries W in column N=0 only; D column 0 lands in
// lanes 0 (M=0..7 over 8 VGPRs) and 16 (M=8..15).
__global__ __launch_bounds__(WAVES_PER_BLOCK * 32)
void gcn_gemv_wmma(const float* __restrict__ x, const float* __restrict__ W,
                   float* __restrict__ h, int N) {
    const int lane = threadIdx.x & 31;
    const int wave = threadIdx.x >> 5;
    const int rowBase = (blockIdx.x * WAVES_PER_BLOCK + wave) * ROWS_PER_WAVE;

    __shared__ float lds[WAVES_PER_BLOCK][ROWS_PER_WAVE][LDS_STRIDE];

    // Coalesced stage: each lane loads one float4 per row (32 lanes * 16B = one 512B row)
    for (int r = 0; r < ROWS_PER_WAVE; ++r) {
        int row = rowBase + r;
        float4 v = make_float4(0.f, 0.f, 0.f, 0.f);
        if (row < N) v = *(const float4*)(x + (size_t)row * D_FEAT + lane * 4);
        lds[wave][r][lane * 4 + 0] = v.x;
        lds[wave][r][lane * 4 + 1] = v.y;
        lds[wave][r][lane * 4 + 2] = v.z;
        lds[wave][r][lane * 4 + 3] = v.w;
    }
    __syncthreads();

    const int half = lane >> 4;      // 0: K={0,1}   1: K={2,3}
    const int m    = lane & 15;      // A: row M;  B/D: column N

    v8f acc = {0.f, 0.f, 0.f, 0.f, 0.f, 0.f, 0.f, 0.f};

    for (int c = 0; c < D_FEAT / 4; ++c) {
        const int k0 = c * 4 + half * 2;
        v2f a;
        a.x = lds[wave][m][k0];
        a.y = lds[wave][m][k0 + 1];
        // B fragment: B[k][0] = W[c*4+k], other columns zero.
        float w0 = W[c * 4 + half * 2 + 0];
        float w1 = W[c * 4 + half * 2 + 1];
        v2f b;
        b.x = (m == 0) ? w0 : 0.f;
        b.y = (m == 0) ? w1 : 0.f;
        acc = __builtin_amdgcn_wmma_f32_16x16x4_f32(
            /*neg_a=*/false, a, /*neg_b=*/false, b,
            /*c_mod=*/(short)0, acc, /*reuse_a=*/false, /*reuse_b=*/false);
    }

    // Write back column N=0 of D.
    if (m == 0) {
        const int mbase = rowBase + half * 8;
#pragma unroll
        for (int j = 0; j < 8; ++j) {
            int row = mbase + j;
            if (row < N) h[row] = acc[j];
        }
    }
}

// ---------------- K3: dinv, g = h*dinv, out seeded with self-loop term -----
__global__ void gcn_node(const float* __restrict__ h, const unsigned* __restrict__ deg,
                         float* __restrict__ dinv, float* __restrict__ g,
                         float* __restrict__ out, int N) {
    int i = blockIdx.x * blockDim.x + threadIdx.x;
    if (i < N) {
        float d  = (float)(deg[i] + 1u);   // +1 self-loop; always >= 1
        float di = rsqrtf(d);
        float gi = h[i] * di;
        dinv[i] = di;
        g[i]    = gi;
        out[i]  = gi * di;                 // self-loop message h[i]*dinv[i]^2
    }
}

// ---------------- K4: edge scatter-add --------------------------------------
__global__ void gcn_scatter(const int* __restrict__ src, const int* __restrict__ dst,
                            const float* __restrict__ g, const float* __restrict__ dinv,
                            float* __restrict__ out, int E) {
    int e = blockIdx.x * blockDim.x + threadIdx.x;
    if (e < E) {
        int s = src[e];
        int d = dst[e];
        atomicAdd(&out[d], g[s] * dinv[d]);
    }
}

// ---------------- K5: + bias, ReLU ------------------------------------------
__global__ void gcn_finalize(float* __restrict__ out, const float* __restrict__ b, int N) {
    int i = blockIdx.x * blockDim.x + threadIdx.x;
    if (i < N) {
        float v = out[i] + b[0];
        out[i] = v > 0.f ? v : 0.f;
    }
}

extern "C" void kernel_launch(void* const* d_in, const int* in_sizes, int n_in,
                              void* d_out, int out_size, void* d_ws, size_t ws_size,
                              hipStream_t stream) {
    const float* x  = (const float*)d_in[0];
    const int*   ei = (const int*)d_in[1];     // [2, E] int32 per harness convention
    const float* W  = (const float*)d_in[2];
    const float* b  = (const float*)d_in[3];

    const int N = in_sizes[0] / D_FEAT;        // 200000
    const int E = in_sizes[1] / 2;             // 6400000
    const int* src = ei;
    const int* dst = ei + E;

    // Workspace layout: [deg u32 | h f32 | dinv f32 | g f32], 4*N words = 3.2 MB
    unsigned* deg = (unsigned*)d_ws;
    float*    h    = (float*)d_ws + (size_t)N;
    float*    dinv = (float*)d_ws + (size_t)2 * N;
    float*    g    = (float*)d_ws + (size_t)3 * N;
    float*    out  = (float*)d_out;

    const int TB = 256;
    gcn_zero_u32<<<(N + TB - 1) / TB, TB, 0, stream>>>(deg, N);
    gcn_degree  <<<(E + TB - 1) / TB, TB, 0, stream>>>(dst, deg, E);

    const int rowsPerBlock = WAVES_PER_BLOCK * ROWS_PER_WAVE;   // 64
    gcn_gemv_wmma<<<(N + rowsPerBlock - 1) / rowsPerBlock, WAVES_PER_BLOCK * 32, 0, stream>>>(x, W, h, N);

    gcn_node    <<<(N + TB - 1) / TB, TB, 0, stream>>>(h, deg, dinv, g, out, N);
    gcn_scatter <<<(E + TB - 1) / TB, TB, 0, stream>>>(src, dst, g, dinv, out, E);
    gcn_finalize<<<(N + TB - 1) / TB, TB, 0, stream>>>(out, b, N);
}